// JCIGBaseline_83004537962758
// MI455X (gfx1250) — compile-verified
//
#include <hip/hip_runtime.h>
#include <hip/hip_bf16.h>

#define NN 100000      // nodes
#define NE 1600000     // edges
#define DI 128         // input dim
#define DH 64          // hidden dim
#define NG 64          // graphs
#define LSTR (DI + 4)  // padded LDS row stride (floats) -> conflict-free b64 reads

typedef __attribute__((ext_vector_type(2))) float v2f;
typedef __attribute__((ext_vector_type(8))) float v8f;

// ---------------- degree / norm ----------------

__global__ void k_init_deg(float* deg) {
    int i = blockIdx.x * blockDim.x + threadIdx.x;
    if (i < NN) deg[i] = 1.0f;   // self-loop weight
}

__global__ void k_accum_deg(float* deg, const int* col, const float* ew) {
    int e = blockIdx.x * blockDim.x + threadIdx.x;
    if (e < NE) atomicAdd(&deg[col[e]], ew[e]);
}

__global__ void k_dinv(float* deg) {
    int i = blockIdx.x * blockDim.x + threadIdx.x;
    if (i < NN) {
        float d = deg[i];
        deg[i] = (d > 0.0f) ? rsqrtf(d) : 0.0f;
    }
}

__global__ void k_norm(float* norm, const float* dinv, const int* row,
                       const int* col, const float* ew) {
    int e = blockIdx.x * blockDim.x + threadIdx.x;
    if (e < NE) norm[e] = dinv[row[e]] * ew[e] * dinv[col[e]];
}

// ---------------- WMMA GEMM + fused self-loop epilogue ----------------
// h[M,64] = A[M,K] @ W[K,64] ; agg[M,64] = dinv[m]^2 * h[M,64]
// One wave owns a 16-row tile x 64 cols via V_WMMA_F32_16X16X4_F32.
// W is staged TRANSPOSED in LDS (sWt[n][k], stride DI+4) so each B fragment
// is one aligned ds_load_b64 straight into a register pair.

__global__ __launch_bounds__(128) void k_gemm_wmma(const float* __restrict__ A,
                                                   const float* __restrict__ W,
                                                   const float* __restrict__ dinv,
                                                   float* __restrict__ h,
                                                   float* __restrict__ agg,
                                                   int M, int K) {
    __shared__ float sWt[DH * LSTR];       // 64 x 132 floats = 33 KB
    const int tid = threadIdx.x;
    const int nW = K * DH;
    for (int i = tid; i < nW; i += 128) {  // W row-major [K,64] -> sWt[n][k]
        int k = i >> 6;                    // i / 64
        int n = i & 63;
        sWt[n * LSTR + k] = W[i];
    }
    __syncthreads();

    const int wave = tid >> 5;
    const int lane = tid & 31;
    const int tile = blockIdx.x * 4 + wave;
    const int numTiles = M >> 4;           // M multiple of 16
    if (tile >= numTiles) return;          // wave-uniform exit (EXEC all-1 below)

    const int m  = lane & 15;              // A row in tile / B-D column
    const int hi = lane >> 4;              // 0 or 1
    const int kb = hi * 2;                 // k sub-offset per ISA A/B layout
    const float* aPtr = A + (long)(tile * 16 + m) * K + kb;
    const float* wPtr = sWt + m * LSTR + kb;   // + nt*16*LSTR per N tile (imm offs)

    v8f c0 = {}, c1 = {}, c2 = {}, c3 = {};
    for (int k = 0; k < K; k += 4) {
        v2f a  = *(const v2f*)(aPtr + k);
        v2f b0 = *(const v2f*)(wPtr + k);
        v2f b1 = *(const v2f*)(wPtr + k + 16 * LSTR);
        v2f b2 = *(const v2f*)(wPtr + k + 32 * LSTR);
        v2f b3 = *(const v2f*)(wPtr + k + 48 * LSTR);
        c0 = __builtin_amdgcn_wmma_f32_16x16x4_f32(false, a, false, b0, (short)0, c0, false, false);
        c1 = __builtin_amdgcn_wmma_f32_16x16x4_f32(false, a, false, b1, (short)0, c1, false, false);
        c2 = __builtin_amdgcn_wmma_f32_16x16x4_f32(false, a, false, b2, (short)0, c2, false, false);
        c3 = __builtin_amdgcn_wmma_f32_16x16x4_f32(false, a, false, b3, (short)0, c3, false, false);
    }

    // C/D layout: VGPR r -> row = tile*16 + 8*hi + r, col = m
    const int row0 = tile * 16 + hi * 8;
    for (int r = 0; r < 8; ++r) {
        const int rr = row0 + r;
        const float dv = dinv[rr];         // broadcast load (uniform per half-wave)
        const float s  = dv * dv;
        const long base = (long)rr * DH + m;
        float v0 = c0[r], v1 = c1[r], v2 = c2[r], v3 = c3[r];
        h[base]        = v0;
        h[base + 16]   = v1;
        h[base + 32]   = v2;
        h[base + 48]   = v3;
        agg[base]      = s * v0;           // self-loop message, fused
        agg[base + 16] = s * v1;
        agg[base + 32] = s * v2;
        agg[base + 48] = s * v3;
    }
}

// ---------------- aggregation ----------------

// agg[col[e],k] += norm[e] * h[row[e],k]   (one thread per edge-feature)
__global__ void k_scatter(float* agg, const float* __restrict__ h,
                          const float* __restrict__ norm,
                          const int* __restrict__ row, const int* __restrict__ col) {
    int idx = blockIdx.x * blockDim.x + threadIdx.x;
    if (idx < NE * DH) {
        int e = idx >> 6;
        int k = idx & 63;
        float v = norm[e] * h[(long)row[e] * DH + k];
        atomicAdd(&agg[(long)col[e] * DH + k], v);
    }
}

// in-place: agg = relu(agg + bias)
__global__ void k_bias_relu(float* agg, const float* bias) {
    int idx = blockIdx.x * blockDim.x + threadIdx.x;
    if (idx < NN * DH) agg[idx] = fmaxf(agg[idx] + bias[idx & 63], 0.0f);
}

// ---------------- pooling + head ----------------

__global__ void k_zero_pool(float* psum, float* cnt) {
    int i = blockIdx.x * blockDim.x + threadIdx.x;
    if (i < NG * DH) psum[i] = 0.0f;
    if (i < NG) cnt[i] = 0.0f;
}

// fused layer-2 bias+relu and segment-sum pooling
__global__ void k_pool(const float* agg, const float* b2, const int* batch,
                       float* psum, float* cnt) {
    int idx = blockIdx.x * blockDim.x + threadIdx.x;
    if (idx < NN * DH) {
        int n = idx >> 6;
        int k = idx & 63;
        float v = fmaxf(agg[idx] + b2[k], 0.0f);
        int g = batch[n];
        atomicAdd(&psum[g * DH + k], v);
        if (k == 0) atomicAdd(&cnt[g], 1.0f);
    }
}

// pooled mean -> Linear(64,32) -> ReLU -> Linear(32,1)
__global__ void k_head(const float* psum, const float* cnt,
                       const float* Wm1, const float* bm1,
                       const float* Wm2, const float* bm2, float* out) {
    int g = threadIdx.x;
    if (g >= NG) return;
    float inv = 1.0f / fmaxf(cnt[g], 1.0f);
    float acc = bm2[0];
    for (int j = 0; j < 32; ++j) {
        float s = bm1[j];
        for (int k = 0; k < DH; ++k)
            s += (psum[g * DH + k] * inv) * Wm1[k * 32 + j];
        acc += fmaxf(s, 0.0f) * Wm2[j];
    }
    out[g] = acc;
}

// ---------------- launch ----------------

extern "C" void kernel_launch(void* const* d_in, const int* in_sizes, int n_in,
                              void* d_out, int out_size, void* d_ws, size_t ws_size,
                              hipStream_t stream) {
    const float* x   = (const float*)d_in[0];
    const int*   ei  = (const int*)  d_in[1];
    const float* ew  = (const float*)d_in[2];
    const int*   b   = (const int*)  d_in[3];
    const float* W1  = (const float*)d_in[4];
    const float* b1  = (const float*)d_in[5];
    const float* W2  = (const float*)d_in[6];
    const float* b2  = (const float*)d_in[7];
    const float* Wm1 = (const float*)d_in[8];
    const float* bm1 = (const float*)d_in[9];
    const float* Wm2 = (const float*)d_in[10];
    const float* bm2 = (const float*)d_in[11];
    float* out = (float*)d_out;

    const int* row = ei;          // ei[0]
    const int* col = ei + NE;     // ei[1]

    float* ws   = (float*)d_ws;
    float* dinv = ws;                       // NN
    float* norm = dinv + NN;                // NE
    float* h    = norm + NE;                // NN*DH
    float* agg  = h + (size_t)NN * DH;      // NN*DH
    float* psum = agg + (size_t)NN * DH;    // NG*DH
    float* cnt  = psum + NG * DH;           // NG

    const int T = 256;
    const int gN   = (NN + T - 1) / T;
    const int gE   = (NE + T - 1) / T;
    const int gNF  = (NN * DH + T - 1) / T;
    const int gEF  = (NE * DH + T - 1) / T;
    const int gTil = (NN / 16 + 3) / 4;     // 4 waves (tiles) per block

    // normalization
    k_init_deg<<<gN, T, 0, stream>>>(dinv);
    k_accum_deg<<<gE, T, 0, stream>>>(dinv, col, ew);
    k_dinv<<<gN, T, 0, stream>>>(dinv);
    k_norm<<<gE, T, 0, stream>>>(norm, dinv, row, col, ew);

    // layer 1 (GEMM fused with self-loop init of agg)
    k_gemm_wmma<<<gTil, 128, 0, stream>>>(x, W1, dinv, h, agg, NN, DI);
    k_scatter<<<gEF, T, 0, stream>>>(agg, h, norm, row, col);
    k_bias_relu<<<gNF, T, 0, stream>>>(agg, b1);

    // layer 2
    k_gemm_wmma<<<gTil, 128, 0, stream>>>(agg, W2, dinv, h, agg, NN, DH);
    k_scatter<<<gEF, T, 0, stream>>>(agg, h, norm, row, col);

    // pool + head
    k_zero_pool<<<(NG * DH + T - 1) / T, T, 0, stream>>>(psum, cnt);
    k_pool<<<gNF, T, 0, stream>>>(agg, b2, b, psum, cnt);
    k_head<<<1, 64, 0, stream>>>(psum, cnt, Wm1, bm1, Wm2, bm2, out);
}